// BatchEKFR_9861244911693
// MI455X (gfx1250) — compile-verified
//
#include <hip/hip_runtime.h>
#include <math.h>
#include <type_traits>

// Problem constants from the reference.
constexpr int T_N  = 128;
constexpr int B_N  = 32;
constexpr int M_N  = 512;
constexpr int H_N  = 64;
constexpr int BM_N = B_N * M_N;   // 16384 neurons; 4 neurons per wave32

__device__ __forceinline__ int   f2i(float x) { return __builtin_bit_cast(int, x); }
__device__ __forceinline__ float i2f(int x)   { return __builtin_bit_cast(float, x); }

// ds_swizzle with compile-time offset (group-of-32 mode).
template<int OFF>
__device__ __forceinline__ float swz(float v) {
    return i2f(__builtin_amdgcn_ds_swizzle(f2i(v), OFF));
}

// v += dpp_shifted(v); CTRL = ROW_SHR:n (0x110|n); invalid lanes contribute 0.
template<int CTRL>
__device__ __forceinline__ float dpp_add(float v) {
    return v + i2f(__builtin_amdgcn_update_dpp(0, f2i(v), CTRL, 0xf, 0xf, true));
}

// Sum over each 8-lane group (lanes 8g..8g+7), result broadcast to all lanes
// of the group.  3 VALU DPP adds accumulate into lanes 7/15/23/31 (shift tree
// never lets cross-group terms reach those lanes), then ONE ds_swizzle
// broadcast: lane' = (lane & 0x18) | 7  ->  offset 0xF8.
__device__ __forceinline__ float group8_sum_bcast(float v) {
    v = dpp_add<0x114>(v);   // row_shr:4
    v = dpp_add<0x112>(v);   // row_shr:2
    v = dpp_add<0x111>(v);   // row_shr:1
    return swz<0xF8>(v);
}

__device__ __forceinline__ float my_tanh(float x) {
#if __has_builtin(__builtin_amdgcn_tanhf)
    return __builtin_amdgcn_tanhf(x);     // gfx1250 v_tanh_f32 (TRANS op)
#else
    return tanhf(x);
#endif
}

__device__ __forceinline__ void ld8(const float* __restrict__ p, float o[8]) {
    const float4 x = *(const float4*)p;
    const float4 y = *(const float4*)(p + 4);
    o[0]=x.x; o[1]=x.y; o[2]=x.z; o[3]=x.w;
    o[4]=y.x; o[5]=y.y; o[6]=y.z; o[7]=y.w;
}

// Compile-time for-loop so DPP ctrl / swizzle offsets / readlane-free selects
// are genuine immediates.
template<int I, int N, typename F>
__device__ __forceinline__ void static_for(F&& f) {
    if constexpr (I < N) {
        f(std::integral_constant<int, I>{});
        static_for<I + 1, N>(f);
    }
}

// ---------------------------------------------------------------------------
// 4 neurons per wave (group = 8 lanes), 8 h-values per lane.
// State u[h] = v[h]*w[h];  z_t = R_t + cur_t*Sa + fs_{t-1}*Sb  with
//   R_t = sum_h d_h*u_{t-1,h}  (group reduction; depends on fs_{t-2}, so it
//   overlaps the previous step's activation),  Sa = sum a*w, Sb = sum 1000*b*w.
// Activation is computed per 8-lane group in VALU (group-uniform values), so
// feedback needs no cross-lane traffic at all; per step there are exactly two
// LDS ops (cur broadcast + R broadcast).
// ---------------------------------------------------------------------------
__global__ __launch_bounds__(256)
void BatchEKFR_neuron_kernel(const float* __restrict__ cur,   // [T,B,M]
                             const float* __restrict__ A,     // [M,H]
                             const float* __restrict__ Bf,    // [M,H]
                             const float* __restrict__ W,     // [M,H]
                             const float* __restrict__ DS,    // [H]
                             const float* __restrict__ GB,    // [M]
                             const float* __restrict__ MC,    // [M]
                             const float* __restrict__ MFR,   // [M]
                             const float* __restrict__ PC,    // [M,4]
                             float* __restrict__ out)         // [T,B,M]
{
    const int tid  = threadIdx.x;
    const int lane = tid & 31;
    const int lig  = lane & 7;                       // lane-in-group
    const int wid  = blockIdx.x * 8 + (tid >> 5);    // global wave id
    const int n    = wid * 4 + (lane >> 3);          // this group's neuron id
    const int m    = n & (M_N - 1);

    // ---- per-lane weights: h = lig*8 + j, j = 0..7 (two b128 loads each) ----
    float av[8], wv[8], bv[8], dv[8];
    ld8(A  + m * H_N + lig * 8, av);
    ld8(W  + m * H_N + lig * 8, wv);
    ld8(Bf + m * H_N + lig * 8, bv);
    ld8(DS + lig * 8, dv);

    float aw[8], bw[8], dk[8];
    float sa = 0.0f, sb = 0.0f;
#pragma unroll
    for (int j = 0; j < 8; ++j) {
        aw[j] = av[j] * wv[j];
        bw[j] = 1000.0f * bv[j] * wv[j];
        dk[j] = 1.0f - dv[j];
        sa += aw[j];
        sb += bw[j];
    }
    const float Sa = group8_sum_bcast(sa);
    const float Sb = group8_sum_bcast(sb);

    // ---- per-neuron scalars (group-uniform, lane resident) ----
    const float gbm    = GB[m];
    const float inv_mc = 1.0f / MC[m];
    const float mfm    = MFR[m];
    const float4 q     = *(const float4*)(PC + m * 4);
    const float c0 = q.x * q.x, c1 = q.y * q.y, c2 = q.z * q.z, c3 = q.w * q.w;

    // ---- preload this neuron's currents: lane lig holds t = lig + 8k ----
    float cl[16];
    const float* __restrict__ cp = cur + n + lig * BM_N;
#pragma unroll
    for (int k = 0; k < 16; ++k) cl[k] = cp[k * 8 * BM_N];  // imm offsets fit

    float* __restrict__ op = out + n + lig * BM_N;

    float u[8], g[8];
#pragma unroll
    for (int j = 0; j < 8; ++j) { u[j] = 0.0f; g[j] = 0.0f; }
    float R = 0.0f, fs = 0.0f, acc = 0.0f;

    static_for<0, T_N>([&](auto tc) {
        constexpr int t = decltype(tc)::value;
        constexpr int j = t & 7;
        constexpr int k = t >> 3;

        // broadcast cur_t within the group: lane' = (lane&0x18) | j
        const float curv = swz<(j << 5) | 0x18>(cl[k]);

        // z_t from group scalars: 2 FMAs off the lagged reduction
        const float z = __builtin_fmaf(curv, Sa, __builtin_fmaf(fs, Sb, R));

        // activation (group-uniform, VALU; covers 4 neurons per instruction)
        const float x    = (z - gbm) * inv_mc;
        const float poly = __builtin_fmaf(
            x, __builtin_fmaf(x, __builtin_fmaf(x, c3, c2), c1), c0);
        const float fsn  = fmaxf(mfm * my_tanh(poly), 0.0f);

        // state update + in-lane partial (independent of activation above)
        float s = 0.0f;
#pragma unroll
        for (int jj = 0; jj < 8; ++jj) {
            u[jj] = __builtin_fmaf(curv, aw[jj], __builtin_fmaf(fs, bw[jj], g[jj]));
            g[jj] = dk[jj] * u[jj];
            s += g[jj];
        }
        R = group8_sum_bcast(s);               // R_{t+1}: 3 DPP + 1 swizzle

        // pack fs into lane j of each group; flush every 8 steps
        acc = (lig == j) ? fsn : acc;
        fs  = fsn;
        if constexpr (j == 7) {
            op[(t - 7) * BM_N] = acc;          // imm offset fits 24-bit
        }
    });

#if __has_builtin(__builtin_amdgcn_s_wait_tensorcnt)
    __builtin_amdgcn_s_wait_tensorcnt(0);
#endif
}

// ---------------------------------------------------------------------------
// Harness entry point.
// ---------------------------------------------------------------------------
extern "C" void kernel_launch(void* const* d_in, const int* in_sizes, int n_in,
                              void* d_out, int out_size, void* d_ws, size_t ws_size,
                              hipStream_t stream) {
    (void)in_sizes; (void)n_in; (void)out_size; (void)d_ws; (void)ws_size;
    const float* cur = (const float*)d_in[0];
    const float* a   = (const float*)d_in[1];
    const float* b   = (const float*)d_in[2];
    const float* w   = (const float*)d_in[3];
    const float* ds  = (const float*)d_in[4];
    const float* gb  = (const float*)d_in[5];
    const float* mc  = (const float*)d_in[6];
    const float* mfr = (const float*)d_in[7];
    const float* pc  = (const float*)d_in[8];
    float* out = (float*)d_out;

    // 16384 neurons, 4 per wave, 8 waves (32 neurons) per block -> 512 blocks.
    dim3 grid(BM_N / 32), block(256);
    hipLaunchKernelGGL(BatchEKFR_neuron_kernel, grid, block, 0, stream,
                       cur, a, b, w, ds, gb, mc, mfr, pc, out);
}